// vector_quantizer_9938554323391
// MI455X (gfx1250) — compile-verified
//
#include <hip/hip_runtime.h>
#include <math.h>

#define B_ROWS  16384
#define K_CODES 4096
#define D_DIM   64
#define DECAYF  0.9f
#define EPSF    1e-5f
#define COMMITF 0.01f

typedef __attribute__((ext_vector_type(2))) float v2f;
typedef __attribute__((ext_vector_type(4))) float v4f;
typedef __attribute__((ext_vector_type(8))) float v8f;

#if __has_builtin(__builtin_amdgcn_global_load_async_to_lds_b128)
#define HAVE_ASYNC_LDS 1
// exact parameter type per hipcc diagnostic: vector_size(16) int, as1 / as3
typedef int v4i_vs __attribute__((vector_size(4 * sizeof(int))));
typedef __attribute__((address_space(1))) v4i_vs* as1_v4i;
typedef __attribute__((address_space(3))) v4i_vs* as3_v4i;
#else
#define HAVE_ASYNC_LDS 0
#endif

// ---- workspace layout (float units) ----
#define DW_OFF      0            // K*D = 262144 floats (scatter accumulation)
#define COUNTS_OFF  262144       // K
#define CSRAW_OFF   266240       // K
#define NORMS_OFF   270336       // K
#define SCAL_OFF    274432       // [0]=n_sum [1]=kl_sum [2]=entropy_sum
#define IDX_OFF     274448       // B ints

// ---- output layout (concatenated tuple, float units) ----
#define OUT_LOSS 0
#define OUT_Q    1
#define OUT_PERP (1 + B_ROWS * D_DIM)            // 1048577
#define OUT_EMB  (OUT_PERP + 1)                  // 1048578
#define OUT_EMAW (OUT_EMB + K_CODES * D_DIM)     // 1310722
#define OUT_CS   (OUT_EMAW + K_CODES * D_DIM)    // 1572866

// -------- zero accumulators (deterministic per launch) --------
__global__ void vq_zero_ws(float* ws) {
  int i = blockIdx.x * 256 + threadIdx.x;
  if (i < K_CODES * D_DIM) ws[DW_OFF + i] = 0.0f;
  if (i < K_CODES)         ws[COUNTS_OFF + i] = 0.0f;
  if (i < 3)               ws[SCAL_OFF + i] = 0.0f;
}

// -------- per-code squared norms --------
__global__ void vq_code_norms(const float* __restrict__ emb, float* ws) {
  int k = blockIdx.x * 256 + threadIdx.x;
  if (k >= K_CODES) return;
  const v4f* e = (const v4f*)(emb + k * D_DIM);
  float s = 0.0f;
#pragma unroll
  for (int i = 0; i < 16; ++i) {
    v4f v = e[i];
    s += v.x * v.x + v.y * v.y + v.z * v.z + v.w * v.w;
  }
  ws[NORMS_OFF + k] = s;
}

// -------- fused distance GEMM (f32 WMMA) + argmin --------
#define KC 128           // codes staged in LDS per chunk
#define LSTR 68          // padded dword stride (64 would be 16-way bank conflict)

__global__ __launch_bounds__(256)
void vq_argmin(const float* __restrict__ x, const float* __restrict__ emb,
               float* ws) {
  __shared__ float smem[KC * LSTR];
  const float* norms = ws + NORMS_OFF;
  int* idx_out = (int*)(ws + IDX_OFF);

  const int tid  = threadIdx.x;
  const int lane = tid & 31;
  const int wave = tid >> 5;
  const int rbase = (blockIdx.x * 8 + wave) * 16;   // 16 rows per wave

  const int mrow = lane & 15;          // A: M index / B: N index for this lane
  const int koff = (lane >> 4) * 2;    // K sub-offset within 4-wide fragment

  // Preload full 16x64 A panel as 16 WMMA f32 16x16x4 A-fragments.
  v2f a[16];
  const float* xrow = x + (size_t)(rbase + mrow) * D_DIM;
#pragma unroll
  for (int f = 0; f < 16; ++f)
    a[f] = *(const v2f*)(xrow + f * 4 + koff);

  float best[8];
  int   bidx[8];
#pragma unroll
  for (int j = 0; j < 8; ++j) { best[j] = 3.4e38f; bidx[j] = 0; }

  for (int c = 0; c < K_CODES / KC; ++c) {
    __syncthreads();
    // ---- stage 128 codebook rows into padded LDS ----
#if HAVE_ASYNC_LDS
#pragma unroll
    for (int j = 0; j < (KC * D_DIM) / (4 * 256); ++j) {   // 8 x b128 per thread
      int e4  = j * 256 + tid;        // float4 index within chunk
      int row = e4 >> 4;              // 16 float4 per code row
      int c4  = e4 & 15;
      as1_v4i g = (as1_v4i)(emb + (size_t)(c * KC + row) * D_DIM + c4 * 4);
      as3_v4i l = (as3_v4i)(&smem[row * LSTR + c4 * 4]);
      __builtin_amdgcn_global_load_async_to_lds_b128(g, l, 0, 0);
    }
#if __has_builtin(__builtin_amdgcn_s_wait_asynccnt)
    __builtin_amdgcn_s_wait_asynccnt(0);
#else
    asm volatile("s_wait_asynccnt 0x0" ::: "memory");
#endif
#else
#pragma unroll
    for (int j = 0; j < (KC * D_DIM) / (4 * 256); ++j) {
      int e4  = j * 256 + tid;
      int row = e4 >> 4;
      int c4  = e4 & 15;
      v4f v = *(const v4f*)(emb + (size_t)(c * KC + row) * D_DIM + c4 * 4);
      *(v4f*)(&smem[row * LSTR + c4 * 4]) = v;
    }
#endif
    __syncthreads();

    // Batch the 8 per-tile code norms for this chunk into registers.
    float nrm[8];
#pragma unroll
    for (int t = 0; t < 8; ++t)
      nrm[t] = norms[c * KC + t * 16 + mrow];

    for (int t = 0; t < KC / 16; ++t) {        // 8 tiles of 16 codes
      const float* bbase = &smem[(t * 16 + mrow) * LSTR + koff];

      // Load all 16 B-fragments first (8x ds_load_2addr_b64, single wait),
      // then fire the WMMAs back-to-back on two independent accumulator chains.
      v2f bf[16];
#pragma unroll
      for (int f = 0; f < 16; ++f)
        bf[f] = *(const v2f*)(bbase + f * 4);

      v8f acc0 = {0.f, 0.f, 0.f, 0.f, 0.f, 0.f, 0.f, 0.f};
      v8f acc1 = {0.f, 0.f, 0.f, 0.f, 0.f, 0.f, 0.f, 0.f};
#pragma unroll
      for (int f = 0; f < 8; ++f) {
        acc0 = __builtin_amdgcn_wmma_f32_16x16x4_f32(
            false, a[2 * f],     false, bf[2 * f],     (short)0, acc0, false, false);
        acc1 = __builtin_amdgcn_wmma_f32_16x16x4_f32(
            false, a[2 * f + 1], false, bf[2 * f + 1], (short)0, acc1, false, false);
      }

      const int nidx = c * KC + t * 16 + mrow;
#pragma unroll
      for (int j = 0; j < 8; ++j) {
        float s = nrm[t] - 2.0f * (acc0[j] + acc1[j]);   // ||e||^2 - 2 x.e
        if (s < best[j]) { best[j] = s; bidx[j] = nidx; }
      }
    }
  }

  // Cross-lane argmin over the 16 columns held per half-wave.
#pragma unroll
  for (int j = 0; j < 8; ++j) {
    float v = best[j];
    int   i = bidx[j];
#pragma unroll
    for (int m = 8; m >= 1; m >>= 1) {
      float ov = __shfl_xor(v, m, 32);
      int   oi = __shfl_xor(i, m, 32);
      if (ov < v || (ov == v && oi < i)) { v = ov; i = oi; }
    }
    if ((lane & 15) == 0)
      idx_out[rbase + j + (lane >> 4) * 8] = i;   // lanes 0..15 -> M=j ; 16..31 -> M=j+8
  }
}

// -------- gather quantized, scatter dw/counts, per-row KL --------
__global__ __launch_bounds__(256)
void vq_gather_kl(const float* __restrict__ x, const float* __restrict__ emb,
                  float* __restrict__ out, float* ws) {
  const int tid  = threadIdx.x;
  const int lane = tid & 31;
  const int wave = tid >> 5;
  const int b = blockIdx.x * 8 + wave;            // one wave per row

  const int* idx = (const int*)(ws + IDX_OFF);
  float* dw     = ws + DW_OFF;
  float* counts = ws + COUNTS_OFF;
  float* scal   = ws + SCAL_OFF;

  const int k = idx[b];
  v2f xv = *(const v2f*)(x + (size_t)b * 64 + 2 * lane);
  v2f qv = *(const v2f*)(emb + (size_t)k * 64 + 2 * lane);
  const float x0 = xv.x, x1 = xv.y;
  const float q0 = qv.x, q1 = qv.y;

  // straight-through forward value == quantized
  *(v2f*)(out + OUT_Q + (size_t)b * 64 + 2 * lane) = qv;

  // target softmax over x (64 elems, wave32: 2 per lane)
  float m = fmaxf(x0, x1);
#pragma unroll
  for (int s = 16; s >= 1; s >>= 1) m = fmaxf(m, __shfl_xor(m, s, 32));
  float e0 = expf(x0 - m), e1 = expf(x1 - m);
  float se = e0 + e1;
#pragma unroll
  for (int s = 16; s >= 1; s >>= 1) se += __shfl_xor(se, s, 32);
  float ls = logf(se);

  // softmax over quantized (used as probabilities, faithful to source)
  float qm = fmaxf(q0, q1);
#pragma unroll
  for (int s = 16; s >= 1; s >>= 1) qm = fmaxf(qm, __shfl_xor(qm, s, 32));
  float f0 = expf(q0 - qm), f1 = expf(q1 - qm);
  float sq = f0 + f1;
#pragma unroll
  for (int s = 16; s >= 1; s >>= 1) sq += __shfl_xor(sq, s, 32);

  float t0 = e0 / se, t1 = e1 / se;
  float kl = t0 * ((x0 - m - ls) - f0 / sq) + t1 * ((x1 - m - ls) - f1 / sq);
#pragma unroll
  for (int s = 16; s >= 1; s >>= 1) kl += __shfl_xor(kl, s, 32);

  atomicAdd(&dw[(size_t)k * 64 + 2 * lane], x0);
  atomicAdd(&dw[(size_t)k * 64 + 2 * lane + 1], x1);
  if (lane == 0) {
    atomicAdd(&counts[k], 1.0f);
    atomicAdd(&scal[1], kl);
  }
}

// -------- cluster-size EMA, global n, entropy --------
__global__ void vq_update_cs(const float* __restrict__ ema_cs, float* ws) {
  int k = blockIdx.x * 256 + threadIdx.x;
  if (k >= K_CODES) return;
  float c  = ws[COUNTS_OFF + k];
  float cr = ema_cs[k] * DECAYF + (1.0f - DECAYF) * c;
  ws[CSRAW_OFF + k] = cr;
  float p = c / (float)B_ROWS;
  atomicAdd(&ws[SCAL_OFF + 0], cr);
  atomicAdd(&ws[SCAL_OFF + 2], -p * logf(p + 1e-10f));
}

// -------- finalize: embedding/ema_w update, cs, loss, perplexity --------
__global__ __launch_bounds__(256)
void vq_finalize(const float* __restrict__ ema_w, float* out, float* ws) {
  int i = blockIdx.x * 256 + threadIdx.x;   // over K*D
  int k = i >> 6;
  int d = i & 63;
  float n  = ws[SCAL_OFF + 0];
  float cr = ws[CSRAW_OFF + k];
  float cs = (cr + EPSF) / (n + (float)K_CODES * EPSF) * n;
  float nw = ema_w[i] * DECAYF + (1.0f - DECAYF) * ws[DW_OFF + i];
  out[OUT_EMAW + i] = nw;
  out[OUT_EMB + i]  = nw / cs;
  if (d == 0) out[OUT_CS + k] = cs;
  if (i == 0) {
    out[OUT_LOSS] = COMMITF * ws[SCAL_OFF + 1] / (float)B_ROWS;
    out[OUT_PERP] = expf(ws[SCAL_OFF + 2]);
  }
}

extern "C" void kernel_launch(void* const* d_in, const int* in_sizes, int n_in,
                              void* d_out, int out_size, void* d_ws, size_t ws_size,
                              hipStream_t stream) {
  const float* x      = (const float*)d_in[0];   // [B, D, 1]
  // d_in[1] = soft [B,1,K] : unused by the reference — never read (saves 256 MB traffic)
  const float* emb    = (const float*)d_in[2];   // [K, D]
  const float* ema_w  = (const float*)d_in[3];   // [K, D]
  const float* ema_cs = (const float*)d_in[4];   // [K]
  float* out = (float*)d_out;
  float* ws  = (float*)d_ws;

  vq_zero_ws   <<<1024, 256, 0, stream>>>(ws);
  vq_code_norms<<<16,   256, 0, stream>>>(emb, ws);
  vq_argmin    <<<B_ROWS / (8 * 16), 256, 0, stream>>>(x, emb, ws);   // 128 WGs
  vq_gather_kl <<<B_ROWS / 8, 256, 0, stream>>>(x, emb, out, ws);     // 2048 WGs
  vq_update_cs <<<16,   256, 0, stream>>>(ema_cs, ws);
  vq_finalize  <<<K_CODES * D_DIM / 256, 256, 0, stream>>>(ema_w, out, ws);
}